// FraudGNN_21492016349642
// MI455X (gfx1250) — compile-verified
//
#include <hip/hip_runtime.h>

typedef __attribute__((ext_vector_type(2))) float v2f;
typedef __attribute__((ext_vector_type(8))) float v8f;

// ---------------------------------------------------------------------------
// CDNA5 async copy: global memory -> LDS, tracked by ASYNCcnt.
// Bypasses the VGPR round-trip (no loadcnt stall, no ds_store).
// ---------------------------------------------------------------------------

__device__ __forceinline__ void async_copy_b128(void* lds_dst, const void* gsrc) {
    __attribute__((address_space(3))) char* l =
        (__attribute__((address_space(3))) char*)lds_dst;
    asm volatile("global_load_async_to_lds_b128 %0, %1, off"
                 :: "v"(l), "v"(gsrc)
                 : "memory");
}

template <int N>
__device__ __forceinline__ void wait_asynccnt() {
    asm volatile("s_wait_asynccnt %0" :: "n"(N) : "memory");
}

// ---------------------------------------------------------------------------
// Utility kernels
// ---------------------------------------------------------------------------

__global__ void zero_f4_kernel(float4* __restrict__ p, long long n4) {
    long long i = (long long)blockIdx.x * blockDim.x + threadIdx.x;
    if (i < n4) p[i] = make_float4(0.f, 0.f, 0.f, 0.f);
}

__global__ void deg_kernel(const long long* __restrict__ ei, float* __restrict__ deg, int E) {
    int e = blockIdx.x * 256 + threadIdx.x;
    if (e < E) atomicAdd(&deg[(long long)(int)ei[(long long)E + e]], 1.0f);
}

__global__ void invdeg_kernel(float* __restrict__ deg, int n) {
    int i = blockIdx.x * 256 + threadIdx.x;
    if (i < n) deg[i] = 1.0f / fmaxf(deg[i], 1.0f);
}

// One edge per (1 << c4shift)-lane slice; each lane moves 4 contiguous
// channels. Gather from h[src] is a 512B coalesced read (L2 resident: the
// whole feature matrix is <= 51.2MB << 192MB L2), scatter is
// global_atomic_add_f32 into agg[dst].
__global__ void scatter_kernel(const float* __restrict__ h, const long long* __restrict__ ei,
                               float* __restrict__ agg, int E, int d, int c4shift) {
    int c4mask = (1 << c4shift) - 1;
    int e = (blockIdx.x << (8 - c4shift)) + (threadIdx.x >> c4shift);
    if (e >= E) return;
    int c = (threadIdx.x & c4mask) << 2;
    int src = (int)ei[e];
    int dst = (int)ei[(long long)E + e];
    float4 v = *(const float4*)(h + (long long)src * d + c);
    float* a = agg + (long long)dst * d + c;
    atomicAdd(a + 0, v.x);
    atomicAdd(a + 1, v.y);
    atomicAdd(a + 2, v.z);
    atomicAdd(a + 3, v.w);
}

// agg[row, :] *= invdeg[row]  (turns the segment-sum into the segment-mean,
// so the GEMM's A-tile staging becomes a pure async copy)
__global__ void scale_rows_kernel(float* __restrict__ agg, const float* __restrict__ invdeg,
                                  long long n4, int rshift) {
    long long i = (long long)blockIdx.x * blockDim.x + threadIdx.x;
    if (i >= n4) return;
    float s = invdeg[(int)(i >> rshift)];
    float4* p = (float4*)agg + i;
    float4 v = *p;
    v.x *= s; v.y *= s; v.z *= s; v.w *= s;
    *p = v;
}

// ---------------------------------------------------------------------------
// Fused SAGE layer GEMM:
//   out = relu( mean @ wl + h @ wr + b )
// One GEMM with A = [mean | h] (N x 2*DIN) and W = [wl ; wr] (2*DIN x DOUT).
// Block = 256 threads (8 wave32). Tile = 128 rows x DOUT cols, BK = 32,
// double-buffered LDS fed by global_load_async_to_lds_b128 with issue-ahead:
// chunk k+1's copies are in flight while chunk k runs WMMAs; the partial
// s_wait_asynccnt <= OPC relies on in-order async-load completion.
// WMMA = V_WMMA_F32_16X16X4_F32 (fp32-exact path: ~10.7 GFLOP total, compute
// is nowhere near the bottleneck, so no bf16 downcast).
// ---------------------------------------------------------------------------

template <int DIN, int DOUT>
__global__ __launch_bounds__(256) void sage_gemm_kernel(
    const float* __restrict__ mean, const float* __restrict__ hin,
    const float* __restrict__ wl, const float* __restrict__ wr,
    const float* __restrict__ bias, float* __restrict__ hout, int n) {

    constexpr int BK  = 32;
    constexpr int K   = 2 * DIN;
    constexpr int NCH = K / BK;          // K-chunks
    constexpr int LDA = 36;              // 144B row stride: 16B-aligned, 16 distinct banks
    constexpr int ASZ = 128 * LDA;       // floats per A buffer
    constexpr int WSZ = BK * DOUT;       // floats per W buffer
    constexpr int NT  = DOUT / 16;       // 16-col tiles per wave
    constexpr int OPC = 4 + DOUT / 32;   // async instructions issued per chunk per wave

    __shared__ float Alds[2 * ASZ];
    __shared__ float Wlds[2 * WSZ];

    const int t    = threadIdx.x;
    const int w    = t >> 5;             // wave id 0..7 -> 16-row stripe
    const int lane = t & 31;
    const int half = lane >> 4;          // K-pair selector (A/B frag layout)
    const int lr   = lane & 15;          // row-in-tile for A, col-in-tile for B/C/D
    const int m0   = blockIdx.x * 128;

    // Accumulators start at the bias: C/D layout fixes the column per lane
    // (col = tile*16 + lr); all 8 vector elements are different rows.
    v8f acc[NT];
#pragma unroll
    for (int nt = 0; nt < NT; ++nt) {
        float bv = bias[nt * 16 + lr];
#pragma unroll
        for (int j = 0; j < 8; ++j) acc[nt][j] = bv;
    }

    auto issue_chunk = [&](int k0, int buf) {
        const bool meanPart = (k0 < DIN);                   // uniform per chunk
        const float* Asrc = meanPart ? (mean + k0) : (hin + (k0 - DIN));
        const float* Wsrc = meanPart ? (wl + (size_t)k0 * DOUT)
                                     : (wr + (size_t)(k0 - DIN) * DOUT);
        float* Abase = &Alds[buf * ASZ];
        float* Wbase = &Wlds[buf * WSZ];
        // A tile: 128 x 32 floats = 1024 float4 -> 4 async B128 per thread
#pragma unroll
        for (int i = 0; i < 4; ++i) {
            int lin = t + i * 256;                          // 0..1023
            int row = lin >> 3;                             // 8 float4 per row
            int kk4 = (lin & 7) << 2;
            int rg = m0 + row;
            rg = rg < n ? rg : n - 1;                       // clamp (no divergence)
            async_copy_b128(&Abase[row * LDA + kk4],
                            Asrc + (long long)rg * DIN + kk4);
        }
        // W tile: 32 x DOUT floats -> DOUT/32 async B128 per thread
#pragma unroll
        for (int i = 0; i < DOUT / 32; ++i) {
            int lin = t + i * 256;                          // < DOUT*8
            int kk  = lin / (DOUT / 4);
            int nn4 = (lin % (DOUT / 4)) << 2;
            async_copy_b128(&Wbase[kk * DOUT + nn4], Wsrc + kk * DOUT + nn4);
        }
    };

    issue_chunk(0, 0);

    for (int c = 0; c < NCH; ++c) {
        const int buf = c & 1;
        if (c + 1 < NCH) {
            // Safe: chunk c-1 (which used buf^1) finished before the trailing
            // barrier of the previous iteration.
            issue_chunk((c + 1) * BK, buf ^ 1);
            wait_asynccnt<OPC>();        // chunk c complete; c+1 still in flight
        } else {
            wait_asynccnt<0>();
        }
        __syncthreads();

        const float* Ab = &Alds[buf * ASZ];
        const float* Wb = &Wlds[buf * WSZ];
#pragma unroll
        for (int kk = 0; kk < BK; kk += 4) {
            // A frag (16x4 f32): lanes 0-15 -> K {kk,kk+1}, lanes 16-31 -> {kk+2,kk+3}
            const float* ap = &Ab[(w * 16 + lr) * LDA + kk + 2 * half];
            v2f a;
            a.x = ap[0];
            a.y = ap[1];
#pragma unroll
            for (int nt = 0; nt < NT; ++nt) {
                int col = nt * 16 + lr;
                v2f b;
                b.x = Wb[(kk + 2 * half + 0) * DOUT + col];
                b.y = Wb[(kk + 2 * half + 1) * DOUT + col];
                acc[nt] = __builtin_amdgcn_wmma_f32_16x16x4_f32(
                    false, a, false, b, (short)0, acc[nt], false, false);
            }
        }
        __syncthreads();
    }

    // Store with ReLU. D layout: lane col = nt*16+lr, rows = 8*half + j.
#pragma unroll
    for (int nt = 0; nt < NT; ++nt) {
#pragma unroll
        for (int j = 0; j < 8; ++j) {
            int row = m0 + w * 16 + half * 8 + j;
            if (row < n) {
                float v = acc[nt][j];
                hout[(long long)row * DOUT + nt * 16 + lr] = v > 0.f ? v : 0.f;
            }
        }
    }
}

// ---------------------------------------------------------------------------
// Classifier: out[i,:] = h3[i,:] @ wc + bc   (32 -> 2)
// ---------------------------------------------------------------------------

__global__ void classifier_kernel(const float* __restrict__ h, const float* __restrict__ wc,
                                  const float* __restrict__ bc, float* __restrict__ out, int n) {
    int i = blockIdx.x * 256 + threadIdx.x;
    if (i >= n) return;
    const float* row = h + (long long)i * 32;
    float o0 = bc[0], o1 = bc[1];
#pragma unroll
    for (int k = 0; k < 32; ++k) {
        float v = row[k];
        o0 += v * wc[2 * k + 0];
        o1 += v * wc[2 * k + 1];
    }
    out[2 * i + 0] = o0;
    out[2 * i + 1] = o1;
}

// ---------------------------------------------------------------------------
// Launch
// ---------------------------------------------------------------------------

static inline size_t align256(size_t x) { return (x + 255) & ~(size_t)255; }

extern "C" void kernel_launch(void* const* d_in, const int* in_sizes, int n_in,
                              void* d_out, int out_size, void* d_ws, size_t ws_size,
                              hipStream_t stream) {
    const float*     x   = (const float*)d_in[0];
    const long long* ei  = (const long long*)d_in[1];
    const float*     w1l = (const float*)d_in[2];
    const float*     b1l = (const float*)d_in[3];
    const float*     w1r = (const float*)d_in[4];
    const float*     w2l = (const float*)d_in[5];
    const float*     b2l = (const float*)d_in[6];
    const float*     w2r = (const float*)d_in[7];
    const float*     w3l = (const float*)d_in[8];
    const float*     b3l = (const float*)d_in[9];
    const float*     w3r = (const float*)d_in[10];
    const float*     wc  = (const float*)d_in[11];
    const float*     bc  = (const float*)d_in[12];
    float*           out = (float*)d_out;

    const int N = in_sizes[0] / 128;
    const int E = in_sizes[1] / 2;

    // Workspace carve-out (invdeg, agg reused per layer, h1, h2; h3 aliases h1)
    char* ws = (char*)d_ws;
    size_t off = 0;
    float* deg = (float*)(ws + off); off += align256((size_t)N * 4);
    float* agg = (float*)(ws + off); off += align256((size_t)N * 128 * 4);
    float* h1  = (float*)(ws + off); off += align256((size_t)N * 128 * 4);
    float* h2  = (float*)(ws + off); off += align256((size_t)N * 64 * 4);
    float* h3  = h1;  // h1 is dead once h2 exists
    (void)ws_size; (void)n_in; (void)out_size;

    const dim3 blk(256);
    auto zero = [&](float* p, long long nf) {
        long long n4 = nf >> 2;
        zero_f4_kernel<<<dim3((unsigned)((n4 + 255) / 256)), blk, 0, stream>>>((float4*)p, n4);
    };

    // Degrees (shared by all three layers: same edge list)
    zero(deg, N);
    deg_kernel<<<dim3((E + 255) / 256), blk, 0, stream>>>(ei, deg, E);
    invdeg_kernel<<<dim3((N + 255) / 256), blk, 0, stream>>>(deg, N);

    const int gemmBlocks = (N + 127) / 128;
    auto scale = [&](float* p, int d, int rshift) {
        long long n4 = (long long)N * (d >> 2);
        scale_rows_kernel<<<dim3((unsigned)((n4 + 255) / 256)), blk, 0, stream>>>(p, deg, n4, rshift);
    };

    // ---- Layer 1: 128 -> 128
    zero(agg, (long long)N * 128);
    scatter_kernel<<<dim3((E + 7) / 8), blk, 0, stream>>>(x, ei, agg, E, 128, 5);
    scale(agg, 128, 5);
    sage_gemm_kernel<128, 128><<<dim3(gemmBlocks), blk, 0, stream>>>(
        agg, x, w1l, w1r, b1l, h1, N);

    // ---- Layer 2: 128 -> 64
    zero(agg, (long long)N * 128);
    scatter_kernel<<<dim3((E + 7) / 8), blk, 0, stream>>>(h1, ei, agg, E, 128, 5);
    scale(agg, 128, 5);
    sage_gemm_kernel<128, 64><<<dim3(gemmBlocks), blk, 0, stream>>>(
        agg, h1, w2l, w2r, b2l, h2, N);

    // ---- Layer 3: 64 -> 32
    zero(agg, (long long)N * 64);
    scatter_kernel<<<dim3((E + 15) / 16), blk, 0, stream>>>(h2, ei, agg, E, 64, 4);
    scale(agg, 64, 4);
    sage_gemm_kernel<64, 32><<<dim3(gemmBlocks), blk, 0, stream>>>(
        agg, h2, w3l, w3r, b3l, h3, N);

    // ---- Classifier: 32 -> 2
    classifier_kernel<<<dim3((N + 255) / 256), blk, 0, stream>>>(h3, wc, bc, out, N);
}